// Seq2SeqBasic_22333829939658
// MI455X (gfx1250) — compile-verified
//
#include <hip/hip_runtime.h>
#include <hip/hip_bf16.h>

typedef __attribute__((ext_vector_type(2))) float v2f;
typedef __attribute__((ext_vector_type(8))) float v8f;

#define S_LEN 512
#define T_LEN 128
#define V_SZ  30000
#define D_SZ  1024
#define HE_SZ 512
#define HD_SZ 1024

__device__ __forceinline__ float sigm(float x) { return 1.0f / (1.0f + expf(-x)); }

// ---------------------------------------------------------------------------
// WMMA f32 GEMM:  C[M,N] = A[M,K] (row-major, stride lda; lda may be negative
// for row reversal) * B[N,K]^T (row-major, stride ldb) + biasA + biasB.
// Requirements (hold for every call site here): M % 16 == 0, N % 16 == 0,
// K % 4 == 0. Block = 128 threads = 4 waves; each wave owns a 16Mx32N tile
// (two 16x16 accumulators sharing one A fragment). The N tail (N % 32 == 16)
// takes a wave-uniform single-tile path -> no per-lane EXEC churn in the
// K loop. Uses V_WMMA_F32_16X16X4_F32; lane layouts per CDNA5 ISA 7.12.2.
// ---------------------------------------------------------------------------
__global__ void wmma_gemm_nt(const float* __restrict__ A, long long lda,
                             const float* __restrict__ B, long long ldb,
                             float* __restrict__ C, long long ldc,
                             int M, int N, int K,
                             const float* __restrict__ biasA,
                             const float* __restrict__ biasB) {
  const int wave = threadIdx.x >> 5;
  const int lane = threadIdx.x & 31;
  const int mBase = blockIdx.y * 16;
  const int nBase = (blockIdx.x * 4 + wave) * 32;
  if (mBase >= M || nBase >= N) return;  // wave-uniform exit (EXEC full for WMMA)

  const int half = lane >> 4;   // which K-pair of the 4-wide K slab this half-wave holds
  const int l16  = lane & 15;
  const int m  = mBase + l16;           // A row held by this lane
  const int n0 = nBase + l16;           // C column, tile 0
  const int n1 = nBase + 16 + l16;      // C column, tile 1
  const bool two = (nBase + 32 <= N);   // wave-uniform: does tile 1 exist?

  const float* aRow  = A + (long long)m * lda;
  const float* bRow0 = B + (long long)n0 * ldb;
  const float* bRow1 = B + (long long)(two ? n1 : n0) * ldb;  // safe ptr, unused if !two

  v8f acc0 = {0.f, 0.f, 0.f, 0.f, 0.f, 0.f, 0.f, 0.f};
  v8f acc1 = {0.f, 0.f, 0.f, 0.f, 0.f, 0.f, 0.f, 0.f};

  if (two) {
#pragma unroll 8
    for (int k = 0; k < K; k += 4) {
      const int kk = k + half * 2;
      v2f av = *(const v2f*)(aRow + kk);    // A[m][kk..kk+1]
      v2f b0 = *(const v2f*)(bRow0 + kk);   // B[kk..kk+1][n0] = W[n0][kk..kk+1]
      v2f b1 = *(const v2f*)(bRow1 + kk);
      acc0 = __builtin_amdgcn_wmma_f32_16x16x4_f32(false, av, false, b0,
                                                   (short)0, acc0, false, false);
      acc1 = __builtin_amdgcn_wmma_f32_16x16x4_f32(false, av, false, b1,
                                                   (short)0, acc1, false, false);
    }
  } else {
#pragma unroll 8
    for (int k = 0; k < K; k += 4) {
      const int kk = k + half * 2;
      v2f av = *(const v2f*)(aRow + kk);
      v2f b0 = *(const v2f*)(bRow0 + kk);
      acc0 = __builtin_amdgcn_wmma_f32_16x16x4_f32(false, av, false, b0,
                                                   (short)0, acc0, false, false);
    }
  }

  float bias0 = 0.f, bias1 = 0.f;
  if (biasA) { bias0 += biasA[n0]; if (two) bias1 += biasA[n1]; }
  if (biasB) { bias0 += biasB[n0]; if (two) bias1 += biasB[n1]; }

  // C/D layout: VGPR r -> row mBase + r (lanes 0-15) / mBase + r + 8 (lanes 16-31)
#pragma unroll
  for (int r = 0; r < 8; ++r) {
    const int row = mBase + r + 8 * half;
    C[(long long)row * ldc + n0] = acc0[r] + bias0;
  }
  if (two) {
#pragma unroll
    for (int r = 0; r < 8; ++r) {
      const int row = mBase + r + 8 * half;
      C[(long long)row * ldc + n1] = acc1[r] + bias1;
    }
  }
}

// ---------------------------------------------------------------------------
// Bidirectional encoder LSTM scan. grid=2 (block 0 = fwd, block 1 = bwd),
// 1024 threads. h lives in LDS; c per-thread for tid<512. 512 steps of
// g = pre[t] + Whh@h (2048x512 matvec, 2 rows/thread) then gates.
// Writes enc[s] = cat(ys_f[s], ys_b_rev[s]) and h0/c0 = cat(f,b).
// ---------------------------------------------------------------------------
__global__ void enc_scan(const float* __restrict__ preF, const float* __restrict__ preB,
                         const float* __restrict__ WhhF, const float* __restrict__ WhhB,
                         float* __restrict__ enc, float* __restrict__ h0,
                         float* __restrict__ c0) {
  const bool bwd = (blockIdx.x == 1);
  const float* pre = bwd ? preB : preF;
  const float* Whh = bwd ? WhhB : WhhF;
  __shared__ float h[HE_SZ];
  __shared__ float g[4 * HE_SZ];
  const int tid = threadIdx.x;  // 1024 threads
  if (tid < HE_SZ) h[tid] = 0.f;
  float c = 0.f;
  __syncthreads();

  const float* w0 = Whh + (long long)tid * HE_SZ;
  const float* w1 = Whh + (long long)(tid + 1024) * HE_SZ;

  for (int t = 0; t < S_LEN; ++t) {
    float acc0 = 0.f, acc1 = 0.f;
#pragma unroll 8
    for (int k = 0; k < HE_SZ; k += 4) {
      float4 hv = *(const float4*)&h[k];
      float4 a  = *(const float4*)&w0[k];
      float4 b  = *(const float4*)&w1[k];
      acc0 += a.x * hv.x + a.y * hv.y + a.z * hv.z + a.w * hv.w;
      acc1 += b.x * hv.x + b.y * hv.y + b.z * hv.z + b.w * hv.w;
    }
    const float* p = pre + (long long)t * (4 * HE_SZ);
    g[tid]        = acc0 + p[tid];
    g[tid + 1024] = acc1 + p[tid + 1024];
    __syncthreads();
    if (tid < HE_SZ) {
      const float gi = g[tid], gf = g[HE_SZ + tid];
      const float gg = g[2 * HE_SZ + tid], go = g[3 * HE_SZ + tid];
      c = sigm(gf) * c + sigm(gi) * tanhf(gg);
      const float hn = sigm(go) * tanhf(c);
      const int row = bwd ? (S_LEN - 1 - t) : t;
      enc[(long long)row * (2 * HE_SZ) + (bwd ? HE_SZ : 0) + tid] = hn;
      h[tid] = hn;
    }
    __syncthreads();
  }
  if (tid < HE_SZ) {
    h0[(bwd ? HE_SZ : 0) + tid] = h[tid];
    c0[(bwd ? HE_SZ : 0) + tid] = c;
  }
}

// ---------------------------------------------------------------------------
// Attention is step-invariant (softmax shift invariance): att = softmax(enc@w_att[D:])
// and context = att @ enc are computed once. One 512-thread block.
// ---------------------------------------------------------------------------
__global__ void attention_ctx(const float* __restrict__ enc,
                              const float* __restrict__ w_att,
                              float* __restrict__ context) {
  __shared__ float wv[D_SZ];
  __shared__ float att[S_LEN];
  __shared__ float red[S_LEN];
  const int tid = threadIdx.x;  // 512
  wv[tid]       = w_att[D_SZ + tid];
  wv[tid + 512] = w_att[D_SZ + tid + 512];
  __syncthreads();

  const float* er = enc + (long long)tid * D_SZ;
  float s = 0.f;
#pragma unroll 8
  for (int k = 0; k < D_SZ; k += 4) {
    float4 e = *(const float4*)&er[k];
    float4 w = *(const float4*)&wv[k];
    s += e.x * w.x + e.y * w.y + e.z * w.z + e.w * w.w;
  }
  red[tid] = s; __syncthreads();
  for (int off = 256; off > 0; off >>= 1) {
    if (tid < off) red[tid] = fmaxf(red[tid], red[tid + off]);
    __syncthreads();
  }
  const float mx = red[0]; __syncthreads();
  const float e = expf(s - mx);
  red[tid] = e; __syncthreads();
  for (int off = 256; off > 0; off >>= 1) {
    if (tid < off) red[tid] += red[tid + off];
    __syncthreads();
  }
  att[tid] = e / red[0]; __syncthreads();

  float c0 = 0.f, c1 = 0.f;
  for (int ss = 0; ss < S_LEN; ++ss) {
    const float a = att[ss];
    c0 += a * enc[(long long)ss * D_SZ + tid];
    c1 += a * enc[(long long)ss * D_SZ + tid + 512];
  }
  context[tid] = c0;
  context[tid + 512] = c1;
}

// dec_in[t] = cat(t==0 ? sos : abstract[t-1],  t==0 ? 0 : context)
__global__ void build_dec_in(const float* __restrict__ abstract,
                             const float* __restrict__ sos,
                             const float* __restrict__ context,
                             float* __restrict__ dec_in) {
  const int i = blockIdx.x * blockDim.x + threadIdx.x;
  if (i >= T_LEN * 2 * D_SZ) return;
  const int t = i >> 11;
  const int c = i & 2047;
  float v;
  if (c < D_SZ) v = (t == 0) ? sos[c] : abstract[(long long)(t - 1) * D_SZ + c];
  else          v = (t == 0) ? 0.f : context[c - D_SZ];
  dec_in[i] = v;
}

// vbias[v] = dot(W_pred[v, D:2D], context) + b_pred[v]; one wave per row.
__global__ void vbias_kernel(const float* __restrict__ W_pred,
                             const float* __restrict__ context,
                             const float* __restrict__ b_pred,
                             float* __restrict__ vbias) {
  const int lane = threadIdx.x & 31;
  const int v = blockIdx.x * 8 + (threadIdx.x >> 5);
  if (v >= V_SZ) return;
  const float* row = W_pred + (long long)v * (2 * D_SZ) + D_SZ;
  float s = 0.f;
#pragma unroll
  for (int i = 0; i < 8; ++i) {
    const int k = i * 128 + lane * 4;
    float4 w = *(const float4*)&row[k];
    float4 c = *(const float4*)&context[k];
    s += w.x * c.x + w.y * c.y + w.z * c.z + w.w * c.w;
  }
  for (int off = 16; off > 0; off >>= 1) s += __shfl_xor(s, off, 32);
  if (lane == 0) vbias[v] = s + b_pred[v];
}

// ---------------------------------------------------------------------------
// Decoder LSTM scan (context folded into dec_pre already). 1 block, 1024 thr.
// Per step: g = dec_pre[t] + Whh_d@h (4096x1024 matvec, 4 rows/thread), gates,
// store h2 to h2_all for the batched logits GEMM.
// ---------------------------------------------------------------------------
__global__ void dec_scan(const float* __restrict__ decPre,
                         const float* __restrict__ WhhD,
                         const float* __restrict__ h0, const float* __restrict__ c0,
                         float* __restrict__ h2_all) {
  __shared__ float h[HD_SZ];
  __shared__ float g[4 * HD_SZ];
  const int tid = threadIdx.x;  // 1024
  h[tid] = h0[tid];
  float c = c0[tid];
  __syncthreads();

  const float* w0 = WhhD + (long long)tid * HD_SZ;
  const float* w1 = w0 + (long long)1024 * HD_SZ;
  const float* w2 = w1 + (long long)1024 * HD_SZ;
  const float* w3 = w2 + (long long)1024 * HD_SZ;

  for (int t = 0; t < T_LEN; ++t) {
    float a0 = 0.f, a1 = 0.f, a2 = 0.f, a3 = 0.f;
#pragma unroll 4
    for (int k = 0; k < HD_SZ; k += 4) {
      float4 hv = *(const float4*)&h[k];
      float4 x0 = *(const float4*)&w0[k];
      float4 x1 = *(const float4*)&w1[k];
      float4 x2 = *(const float4*)&w2[k];
      float4 x3 = *(const float4*)&w3[k];
      a0 += x0.x * hv.x + x0.y * hv.y + x0.z * hv.z + x0.w * hv.w;
      a1 += x1.x * hv.x + x1.y * hv.y + x1.z * hv.z + x1.w * hv.w;
      a2 += x2.x * hv.x + x2.y * hv.y + x2.z * hv.z + x2.w * hv.w;
      a3 += x3.x * hv.x + x3.y * hv.y + x3.z * hv.z + x3.w * hv.w;
    }
    const float* p = decPre + (long long)t * (4 * HD_SZ);
    g[tid]        = a0 + p[tid];
    g[tid + 1024] = a1 + p[tid + 1024];
    g[tid + 2048] = a2 + p[tid + 2048];
    g[tid + 3072] = a3 + p[tid + 3072];
    __syncthreads();
    const float gi = g[tid], gf = g[tid + 1024], gg = g[tid + 2048], go = g[tid + 3072];
    c = sigm(gf) * c + sigm(gi) * tanhf(gg);
    const float hn = sigm(go) * tanhf(c);
    __syncthreads();
    h[tid] = hn;
    h2_all[(long long)t * HD_SZ + tid] = hn;
    __syncthreads();
  }
}

__global__ void zero_out(float* out) { out[0] = 0.f; }

// loss += (logsumexp(logits[t]) - logits[t][label_t]) / T ; one block per row.
__global__ void loss_kernel(const float* __restrict__ logits,
                            const int* __restrict__ labels,
                            float* __restrict__ out) {
  __shared__ float red[256];
  const int t = blockIdx.x;
  const int tid = threadIdx.x;
  const float* row = logits + (long long)t * V_SZ;
  float mx = -3.402823466e38f;
  for (int v = tid; v < V_SZ; v += 256) mx = fmaxf(mx, row[v]);
  red[tid] = mx; __syncthreads();
  for (int off = 128; off > 0; off >>= 1) {
    if (tid < off) red[tid] = fmaxf(red[tid], red[tid + off]);
    __syncthreads();
  }
  mx = red[0]; __syncthreads();
  float s = 0.f;
  for (int v = tid; v < V_SZ; v += 256) s += expf(row[v] - mx);
  red[tid] = s; __syncthreads();
  for (int off = 128; off > 0; off >>= 1) {
    if (tid < off) red[tid] += red[tid + off];
    __syncthreads();
  }
  if (tid == 0) {
    const float lse = logf(red[0]) + mx;
    atomicAdd(out, (lse - row[labels[t]]) * (1.0f / (float)T_LEN));
  }
}

// ---------------------------------------------------------------------------
extern "C" void kernel_launch(void* const* d_in, const int* in_sizes, int n_in,
                              void* d_out, int out_size, void* d_ws, size_t ws_size,
                              hipStream_t stream) {
  const float* article  = (const float*)d_in[0];
  const float* abstract = (const float*)d_in[1];
  const int*   labels   = (const int*)d_in[2];
  const float* sos      = (const float*)d_in[3];
  const float* Wih_f    = (const float*)d_in[4];
  const float* Whh_f    = (const float*)d_in[5];
  const float* bih_f    = (const float*)d_in[6];
  const float* bhh_f    = (const float*)d_in[7];
  const float* Wih_b    = (const float*)d_in[8];
  const float* Whh_b    = (const float*)d_in[9];
  const float* bih_b    = (const float*)d_in[10];
  const float* bhh_b    = (const float*)d_in[11];
  const float* Wih_d    = (const float*)d_in[12];
  const float* Whh_d    = (const float*)d_in[13];
  const float* bih_d    = (const float*)d_in[14];
  const float* bhh_d    = (const float*)d_in[15];
  const float* w_att    = (const float*)d_in[16];
  const float* W_pred   = (const float*)d_in[18];
  const float* b_pred   = (const float*)d_in[19];
  float* out = (float*)d_out;

  float* ws = (float*)d_ws;
  float* pre_f   = ws;                       // 512*2048
  float* pre_b   = pre_f   + 512 * 2048;     // 512*2048
  float* enc     = pre_b   + 512 * 2048;     // 512*1024
  float* h0      = enc     + 512 * 1024;     // 1024
  float* c0      = h0      + 1024;           // 1024
  float* context = c0      + 1024;           // 1024
  float* dec_in  = context + 1024;           // 128*2048
  float* dec_pre = dec_in  + 128 * 2048;     // 128*4096
  float* vbias   = dec_pre + 128 * 4096;     // 30000
  float* h2_all  = vbias   + 30000;          // 128*1024
  float* logits  = h2_all  + 128 * 1024;     // 128*30000

  dim3 blk(128);

  // 1) Encoder input projections (hoisted out of the scan): WMMA GEMMs.
  {
    dim3 grd((4 * HE_SZ + 127) / 128, S_LEN / 16);
    wmma_gemm_nt<<<grd, blk, 0, stream>>>(article, D_SZ, Wih_f, D_SZ,
                                          pre_f, 4 * HE_SZ, S_LEN, 4 * HE_SZ, D_SZ,
                                          bih_f, bhh_f);
    // backward direction: reversed rows of x via negative lda
    wmma_gemm_nt<<<grd, blk, 0, stream>>>(article + (long long)(S_LEN - 1) * D_SZ,
                                          -(long long)D_SZ, Wih_b, D_SZ,
                                          pre_b, 4 * HE_SZ, S_LEN, 4 * HE_SZ, D_SZ,
                                          bih_b, bhh_b);
  }

  // 2) Bidirectional encoder recurrence (latency-bound; 2 persistent WGs).
  enc_scan<<<2, 1024, 0, stream>>>(pre_f, pre_b, Whh_f, Whh_b, enc, h0, c0);

  // 3) Step-invariant attention context (softmax shift-invariance).
  attention_ctx<<<1, 512, 0, stream>>>(enc, w_att, context);

  // 4) Decoder input projection over all steps (context is constant).
  build_dec_in<<<(T_LEN * 2 * D_SZ + 255) / 256, 256, 0, stream>>>(abstract, sos,
                                                                   context, dec_in);
  {
    dim3 grd((4 * HD_SZ + 127) / 128, T_LEN / 16);
    wmma_gemm_nt<<<grd, blk, 0, stream>>>(dec_in, 2 * D_SZ, Wih_d, 2 * D_SZ,
                                          dec_pre, 4 * HD_SZ, T_LEN, 4 * HD_SZ,
                                          2 * D_SZ, bih_d, bhh_d);
  }

  // 5) Constant half of the prediction: vbias = W_pred[:,D:] @ context + b_pred.
  vbias_kernel<<<(V_SZ + 7) / 8, 256, 0, stream>>>(W_pred, context, b_pred, vbias);

  // 6) Decoder recurrence; stores all h2 for the batched logits GEMM.
  dec_scan<<<1, 1024, 0, stream>>>(dec_pre, Whh_d, h0, c0, h2_all);

  // 7) Batched logits GEMM: [128x1024] x W_pred[:, :D]^T  (+ vbias).
  //    Bandwidth-bound on W_pred (123 MB) -> f32 WMMA keeps exact precision.
  {
    dim3 grd((V_SZ + 127) / 128, T_LEN / 16);
    wmma_gemm_nt<<<grd, blk, 0, stream>>>(h2_all, HD_SZ, W_pred, 2 * D_SZ,
                                          logits, V_SZ, T_LEN, V_SZ, HD_SZ,
                                          vbias, nullptr);
  }

  // 8) Cross-entropy loss reduction.
  zero_out<<<1, 1, 0, stream>>>(out);
  loss_kernel<<<T_LEN, 256, 0, stream>>>(logits, labels, out);
}